// CTCGreedySearch_7756710937360
// MI455X (gfx1250) — compile-verified
//
#include <hip/hip_runtime.h>
#include <stdint.h>

#define T_DIM 2000
#define N_DIM 64
#define V_DIM 1000
#define VEC4  (V_DIM / 4)        // 250 float4 per row
#define WAVES 8                  // waves per block in kernel 1
#define BLANK 0

// ---------------------------------------------------------------------------
// Kernel 1: per-(t,n)-row fused max / argmax / logsumexp.
// One wave32 per row. Row data staged into LDS via CDNA5 async global->LDS
// loads (ASYNCcnt-tracked, non-temporal: 512MB stream > 192MB L2), then read
// twice from LDS.
// ---------------------------------------------------------------------------
__global__ __launch_bounds__(WAVES * 32) void ctc_row_reduce(
    const float* __restrict__ logits,  // (T, N, V) flat; row r = t*N + n
    int* __restrict__ ws_arg,          // (N, T) argmax
    float* __restrict__ ws_max)        // (N, T) max of log_softmax
{
    __shared__ float4 tile[WAVES][VEC4];   // 4000 B per wave

    const int lane = threadIdx.x & 31;
    const int wv   = threadIdx.x >> 5;
    const int r    = blockIdx.x * WAVES + wv;          // row index, < T*N exactly
    const unsigned rowByte = (unsigned)r * (unsigned)(V_DIM * sizeof(float));
    const unsigned long long sbase = (unsigned long long)(uintptr_t)logits;

    // ---- issue async loads: lane handles float4 indices lane, lane+32, ... ----
#pragma unroll
    for (int k = 0; k < 8; ++k) {
        const int i = lane + k * 32;
        if (i < VEC4) {
            // low 32 bits of a __shared__ generic pointer == workgroup-relative LDS byte addr
            const unsigned ldsAddr = (unsigned)(uintptr_t)(&tile[wv][i]);
            const unsigned voff    = rowByte + (unsigned)i * 16u;
            asm volatile("global_load_async_to_lds_b128 %0, %1, %2 th:TH_LOAD_NT"
                         :: "v"(ldsAddr), "v"(voff), "s"(sbase)
                         : "memory");
        }
    }
    asm volatile("s_wait_asynccnt 0" ::: "memory");

    // ---- pass 1: max + argmax (first occurrence wins) ----
    float m  = -__builtin_inff();
    int   am = 0;
#pragma unroll
    for (int k = 0; k < 8; ++k) {
        const int i = lane + k * 32;
        if (i < VEC4) {
            const float4 v = tile[wv][i];
            const int s = i * 4;
            if (v.x > m) { m = v.x; am = s;     }
            if (v.y > m) { m = v.y; am = s + 1; }
            if (v.z > m) { m = v.z; am = s + 2; }
            if (v.w > m) { m = v.w; am = s + 3; }
        }
    }
#pragma unroll
    for (int off = 16; off > 0; off >>= 1) {
        const float om = __shfl_xor(m, off, 32);
        const int   oa = __shfl_xor(am, off, 32);
        if (om > m || (om == m && oa < am)) { m = om; am = oa; }
    }

    // ---- pass 2: sum of exp(x - m) ----
    float s = 0.f;
#pragma unroll
    for (int k = 0; k < 8; ++k) {
        const int i = lane + k * 32;
        if (i < VEC4) {
            const float4 v = tile[wv][i];
            s += __expf(v.x - m) + __expf(v.y - m) +
                 __expf(v.z - m) + __expf(v.w - m);
        }
    }
#pragma unroll
    for (int off = 16; off > 0; off >>= 1) s += __shfl_xor(s, off, 32);

    if (lane == 0) {
        const int n = r & (N_DIM - 1);
        const int t = r >> 6;                 // r / N_DIM (N_DIM == 64)
        ws_arg[n * T_DIM + t] = am;
        ws_max[n * T_DIM + t] = -__logf(s);  // max(log_softmax) = -log(sum exp(x-m))
    }
}

// ---------------------------------------------------------------------------
// Kernel 2: per-batch-element scan: blank/dup removal, stream compaction,
// masked score sum. One wave32 per n, chunked wave prefix scan over T.
// ---------------------------------------------------------------------------
__global__ __launch_bounds__(32) void ctc_compact(
    const int* __restrict__ ws_arg,
    const float* __restrict__ ws_max,
    const int* __restrict__ in_lens,
    float* __restrict__ out)               // [score N | paths T*N | out_lens N]
{
    __shared__ int lds_arg[T_DIM];
    __shared__ int lds_comp[T_DIM];

    const int n    = blockIdx.x;
    const int lane = threadIdx.x;
    const int L    = in_lens[n];

    int   carry_prev = -1;   // argmax never negative -> no dup-kill at t==0
    int   base       = 0;    // kept count so far
    float sc         = 0.f;

    constexpr int CHUNKS = (T_DIM + 31) / 32;
#pragma unroll 1
    for (int c = 0; c < CHUNKS; ++c) {
        const int  t     = c * 32 + lane;
        const bool valid = t < T_DIM;

        const int   a = valid ? ws_arg[n * T_DIM + t] : -1;
        const float v = (valid && t < L) ? ws_max[n * T_DIM + t] : 0.f;
        sc += v;

        int prev = __shfl_up(a, 1, 32);
        if (lane == 0) prev = carry_prev;

        const int keep = (valid && t < L && a != BLANK && a != prev) ? 1 : 0;

        int scan = keep;                      // inclusive prefix sum over lanes
#pragma unroll
        for (int off = 1; off < 32; off <<= 1) {
            const int x = __shfl_up(scan, off, 32);
            if (lane >= off) scan += x;
        }

        if (valid) lds_arg[t] = a;
        if (keep)  lds_comp[base + scan - 1] = a;

        base       += __shfl(scan, 31, 32);
        carry_prev  = __shfl(a, 31, 32);
    }

#pragma unroll
    for (int off = 16; off > 0; off >>= 1) sc += __shfl_xor(sc, off, 32);

    if (lane == 0) {
        out[n] = sc;                                   // score
        out[N_DIM + T_DIM * N_DIM + n] = (float)base;  // out_lens
    }

    // paths (T, N): compacted inside out_len, original argmax outside.
    // Never re-read on device -> non-temporal stores.
#pragma unroll 1
    for (int c = 0; c < CHUNKS; ++c) {
        const int t = c * 32 + lane;
        if (t < T_DIM) {
            const int val = (t < base) ? lds_comp[t] : lds_arg[t];
            __builtin_nontemporal_store((float)val, &out[N_DIM + t * N_DIM + n]);
        }
    }
}

// ---------------------------------------------------------------------------
extern "C" void kernel_launch(void* const* d_in, const int* in_sizes, int n_in,
                              void* d_out, int out_size, void* d_ws, size_t ws_size,
                              hipStream_t stream)
{
    const float* logits  = (const float*)d_in[0];
    const int*   in_lens = (const int*)d_in[1];
    float*       out     = (float*)d_out;

    int*   ws_arg = (int*)d_ws;                                   // N*T ints
    float* ws_max = (float*)((char*)d_ws + sizeof(int) * (size_t)N_DIM * T_DIM);

    const int rows   = T_DIM * N_DIM;          // 128000
    const int blocks = rows / WAVES;           // 16000

    ctc_row_reduce<<<blocks, WAVES * 32, 0, stream>>>(logits, ws_arg, ws_max);
    ctc_compact<<<N_DIM, 32, 0, stream>>>(ws_arg, ws_max, in_lens, out);
}